// Weighted4DLUTInter_7335804141899
// MI455X (gfx1250) — compile-verified
//
#include <hip/hip_runtime.h>
#include <stdint.h>

// Problem constants (match reference)
#define LN   4          // number of LUTs
#define DD   17         // lattice size per dim
#define SS   4          // upsample factor
#define BB   4
#define HH   256
#define WW   256
#define D4   (17*17*17*17)   // 83521 entries per LUT

typedef unsigned int v4u __attribute__((ext_vector_type(4)));
typedef int          v4i __attribute__((ext_vector_type(4)));
typedef int          v8i __attribute__((ext_vector_type(8)));

#if defined(__has_builtin)
#  if __has_builtin(__builtin_amdgcn_tensor_load_to_lds)
#    define HAS_TDM 1
#  else
#    define HAS_TDM 0
#  endif
#else
#  define HAS_TDM 0
#endif

#if HAS_TDM
// Issue one 2-D TDM tile load: tile (tile_d0 x tile_d1) elements of 4B,
// rows separated by d0_stride elements, global->LDS. Descriptor layout per
// CDNA5 ISA ch. 7/8 (async_tensor): group0 = {count,lds_addr,global_addr,type},
// group1 = {data_size, tensor dims, tile dims, strides}.
__device__ __forceinline__ void tdm_load_2d(uint32_t lds_off, uint64_t gaddr,
                                            uint32_t tensor_d0, uint32_t tensor_d1,
                                            uint32_t tile_d0, uint32_t tile_d1,
                                            uint64_t d0_stride)
{
    v4u g0;
    g0[0] = 1u;                                   // count=1, user desc, no gather
    g0[1] = lds_off;                              // LDS byte address
    g0[2] = (uint32_t)gaddr;                      // global_addr[31:0]
    g0[3] = (uint32_t)(gaddr >> 32) | 0x80000000u;// global_addr[56:32] | type=2

    v8i g1;
    g1[0] = (int)(2u << 16);                      // wg_mask=0, data_size=2 (4B)
    g1[1] = (int)((tensor_d0 & 0xFFFFu) << 16);   // atomic_addr=0 | dim0[15:0]
    g1[2] = (int)(((tensor_d0 >> 16) & 0xFFFFu) | ((tensor_d1 & 0xFFFFu) << 16));
    g1[3] = (int)(((tensor_d1 >> 16) & 0xFFFFu) | ((tile_d0 & 0xFFFFu) << 16));
    g1[4] = (int)(tile_d1 & 0xFFFFu);             // tile_dim1 | tile_dim2=0
    g1[5] = (int)(uint32_t)d0_stride;             // dim0_stride[31:0]
    g1[6] = (int)((d0_stride >> 32) & 0xFFFFu);   // dim0_stride[47:32] | dim1_stride lo = 0
    g1[7] = 0;

    v4i z4 = {0, 0, 0, 0};
#if __has_include(<hip/amd_detail/amd_gfx1250_TDM.h>)
    v8i z8 = {0, 0, 0, 0, 0, 0, 0, 0};
    __builtin_amdgcn_tensor_load_to_lds(g0, g1, z4, z4, z8, 0);   // clang-23 6-arg
#else
    __builtin_amdgcn_tensor_load_to_lds(g0, g1, z4, z4, 0);       // ROCm 7.2 5-arg
#endif
}
#endif // HAS_TDM

__device__ __forceinline__ uint32_t lds_off32(const void* p) {
    // Low 32 bits of a generic LDS address are the LDS byte offset (ISA 10.2).
    return (uint32_t)(uintptr_t)p;
}

__device__ __forceinline__ int dimstride(int i) {
    // strides of dims 0..3 in the 17^4 lattice
    return i == 0 ? (DD*DD*DD) : (i == 1 ? (DD*DD) : (i == 2 ? DD : 1));
}

__global__ __launch_bounds__(WW) void lut4d_kernel(
    const float* __restrict__ lut,      // (L, 17^4, 16)
    const float* __restrict__ weight,   // (B, L, H, W)
    const float* __restrict__ x,        // (B, 1, H, W)
    float* __restrict__ out)            // (B, 1, H*S, W*S)
{
    __shared__ float s_w[LN][WW];
    __shared__ float s_x0[WW];
    __shared__ float s_x1[WW];

    const int bid = blockIdx.x;       // one block per image row
    const int b = bid / HH;
    const int h = bid % HH;
    const int w = threadIdx.x;
    const int hn = (h + 1 < HH) ? h + 1 : HH - 2;   // reflect pad bottom

#if HAS_TDM
    if (threadIdx.x < 32) {           // wave 0 drives the Tensor Data Mover
        uint64_t gw  = (uint64_t)(uintptr_t)(weight + ((size_t)b * LN * HH + h) * WW);
        tdm_load_2d(lds_off32(&s_w[0][0]), gw, WW, LN, WW, LN, (uint64_t)(HH * WW));
        uint64_t gx0 = (uint64_t)(uintptr_t)(x + ((size_t)b * HH + h) * WW);
        tdm_load_2d(lds_off32(&s_x0[0]), gx0, WW, 1, WW, 1, (uint64_t)WW);
        uint64_t gx1 = (uint64_t)(uintptr_t)(x + ((size_t)b * HH + hn) * WW);
        tdm_load_2d(lds_off32(&s_x1[0]), gx1, WW, 1, WW, 1, (uint64_t)WW);
        __builtin_amdgcn_s_wait_tensorcnt(0);
    }
#else
    // Fallback staging with plain coalesced loads
    {
        const float* wr = weight + ((size_t)b * LN * HH + h) * WW;
        #pragma unroll
        for (int l = 0; l < LN; ++l) s_w[l][w] = wr[(size_t)l * HH * WW + w];
        s_x0[w] = x[((size_t)b * HH + h)  * WW + w];
        s_x1[w] = x[((size_t)b * HH + hn) * WW + w];
    }
#endif
    __syncthreads();

    // 2x2 neighborhood (reflect pad right)
    const int w2 = (w + 1 < WW) ? w + 1 : WW - 2;
    float p0 = s_x0[w], p1 = s_x0[w2], p2 = s_x1[w], p3 = s_x1[w2];

    float fr[4]; int bs[4];
    float pv[4] = {p0, p1, p2, p3};
    #pragma unroll
    for (int i = 0; i < 4; ++i) {
        float t  = pv[i] * (1.0f / 16.0f);
        float ff = floorf(t);
        fr[i] = t - ff;
        int bi = (int)ff;
        bi = bi < 0 ? 0 : (bi > DD - 2 ? DD - 2 : bi);
        bs[i] = bi;
    }
    int flat0 = ((bs[0] * DD + bs[1]) * DD + bs[2]) * DD + bs[3];

    // Descending sort of fractions with dim-index tracking (5-comparator network)
    float f0 = fr[0], f1 = fr[1], f2 = fr[2], f3 = fr[3];
    int   i0 = 0,     i1 = 1,     i2 = 2,     i3 = 3;
#define CSWP(fa, fb, ia, ib) \
    if (fb > fa) { float tf = fa; fa = fb; fb = tf; int ti = ia; ia = ib; ib = ti; }
    CSWP(f0, f1, i0, i1)
    CSWP(f2, f3, i2, i3)
    CSWP(f0, f2, i0, i2)
    CSWP(f1, f3, i1, i3)
    CSWP(f1, f2, i1, i2)
#undef CSWP

    float wt[5] = {1.0f - f0, f0 - f1, f1 - f2, f2 - f3, f3};
    int fl[5];
    fl[0] = flat0;
    fl[1] = fl[0] + dimstride(i0);
    fl[2] = fl[1] + dimstride(i1);
    fl[3] = fl[2] + dimstride(i2);
    fl[4] = fl[3] + dimstride(i3);

    // Gather-accumulate: 20 x 64B cacheline gathers, 320 FMAs
    float4 acc[SS];
    #pragma unroll
    for (int si = 0; si < SS; ++si) acc[si] = make_float4(0.f, 0.f, 0.f, 0.f);

    const float4* lut4 = (const float4*)lut;
    #pragma unroll
    for (int l = 0; l < LN; ++l) {
        float wl = s_w[l][w];
        #pragma unroll
        for (int k = 0; k < 5; ++k) {
            float c = wl * wt[k];
            const float4* e = lut4 + ((size_t)l * D4 + (size_t)fl[k]) * 4;
            float4 e0 = e[0], e1 = e[1], e2 = e[2], e3 = e[3];
            acc[0].x = fmaf(c, e0.x, acc[0].x); acc[0].y = fmaf(c, e0.y, acc[0].y);
            acc[0].z = fmaf(c, e0.z, acc[0].z); acc[0].w = fmaf(c, e0.w, acc[0].w);
            acc[1].x = fmaf(c, e1.x, acc[1].x); acc[1].y = fmaf(c, e1.y, acc[1].y);
            acc[1].z = fmaf(c, e1.z, acc[1].z); acc[1].w = fmaf(c, e1.w, acc[1].w);
            acc[2].x = fmaf(c, e2.x, acc[2].x); acc[2].y = fmaf(c, e2.y, acc[2].y);
            acc[2].z = fmaf(c, e2.z, acc[2].z); acc[2].w = fmaf(c, e2.w, acc[2].w);
            acc[3].x = fmaf(c, e3.x, acc[3].x); acc[3].y = fmaf(c, e3.y, acc[3].y);
            acc[3].z = fmaf(c, e3.z, acc[3].z); acc[3].w = fmaf(c, e3.w, acc[3].w);
        }
    }

    // Write 4x4 patch: out[b, 0, h*4+si, w*4 + 0..3], coalesced b128 stores
    float4* out4 = (float4*)out;
    size_t rowbase = (size_t)b * (HH * SS) + (size_t)h * SS;
    #pragma unroll
    for (int si = 0; si < SS; ++si) {
        out4[(rowbase + si) * WW + w] = acc[si];   // row = 1024 floats = 256 float4
    }
}

extern "C" void kernel_launch(void* const* d_in, const int* in_sizes, int n_in,
                              void* d_out, int out_size, void* d_ws, size_t ws_size,
                              hipStream_t stream) {
    // Inputs (reference order): lut, tri_index (unused), weight, x, scale_factor (compile-time 4)
    const float* lut    = (const float*)d_in[0];
    const float* weight = (const float*)d_in[2];
    const float* x      = (const float*)d_in[3];
    float* out          = (float*)d_out;

    (void)in_sizes; (void)n_in; (void)d_ws; (void)ws_size; (void)out_size;

    lut4d_kernel<<<dim3(BB * HH), dim3(WW), 0, stream>>>(lut, weight, x, out);
}